// AttentionalGraphInteractLayer_22909355556939
// MI455X (gfx1250) — compile-verified
//
#include <hip/hip_runtime.h>
#include <hip/hip_bf16.h>
#include <math.h>

#define Nn 8192
#define Dd 1024
#define Oo 1024

typedef __attribute__((ext_vector_type(2))) float v2f;
typedef __attribute__((ext_vector_type(4))) float v4f;
typedef __attribute__((ext_vector_type(8))) float v8f;

__device__ __forceinline__ v8f zero8() {
    v8f z;
    #pragma unroll
    for (int i = 0; i < 8; ++i) z[i] = 0.0f;
    return z;
}

// D = A(16x4) * B(4x16) + C, fp32 WMMA (CDNA5)
__device__ __forceinline__ v8f wmma4(v2f a, v2f b, v8f c) {
    return __builtin_amdgcn_wmma_f32_16x16x4_f32(
        /*neg_a=*/false, a, /*neg_b=*/false, b,
        /*c_mod=*/(short)0, c, /*reuse_a=*/false, /*reuse_b=*/false);
}

// ---------------------------------------------------------------------------
// 64x64-tiled transpose: out[c][r] = in[r][c].  R, C multiples of 64.
// ---------------------------------------------------------------------------
__global__ __launch_bounds__(256)
void transpose_kernel(const float* __restrict__ in, float* __restrict__ out,
                      int R, int C) {
    __shared__ float tile[64][65];
    const int bx = blockIdx.x * 64;          // input col base
    const int by = blockIdx.y * 64;          // input row base
    const int tc = threadIdx.x & 63;
    const int tr = threadIdx.x >> 6;         // 0..3
    #pragma unroll
    for (int i = 0; i < 64; i += 4)
        tile[tr + i][tc] = in[(size_t)(by + tr + i) * C + bx + tc];
    __syncthreads();
    #pragma unroll
    for (int i = 0; i < 64; i += 4)
        out[(size_t)(bx + tr + i) * R + by + tc] = tile[tc][tr + i];
}

// ---------------------------------------------------------------------------
// NT GEMM: C = A[Nrows,K] @ BT[M,K]^T + bias[M]
// Block = 256 threads (8 waves); wave w -> rows [blk.x*128+w*16,+16),
// cols [blk.y*64,+64).  A and B fragments are both contiguous along K
// (single global_load_b64 each).  TOUT=1 writes C transposed ([M,Nrows])
// with vectorized b128 stores.
// ---------------------------------------------------------------------------
template <bool TOUT>
__global__ __launch_bounds__(256)
void gemm_nt_bias_kernel(const float* __restrict__ A, const float* __restrict__ BT,
                         const float* __restrict__ bias, float* __restrict__ C,
                         int Nrows, int K, int M) {
    const int wave = threadIdx.x >> 5;
    const int lane = threadIdx.x & 31;
    const int h    = lane >> 4;              // half-wave K split of fragment
    const int r16  = lane & 15;
    const int m0   = blockIdx.x * 128 + wave * 16;
    const int c0   = blockIdx.y * 64;

    v8f acc[4];
    #pragma unroll
    for (int t = 0; t < 4; ++t) acc[t] = zero8();

    const float* Arow = A + (size_t)(m0 + r16) * K;
    const float* Brow[4];
    #pragma unroll
    for (int t = 0; t < 4; ++t) Brow[t] = BT + (size_t)(c0 + t * 16 + r16) * K;

    for (int k = 0; k < K; k += 4) {
        v2f a = *(const v2f*)(Arow + k + 2 * h);
        #pragma unroll
        for (int t = 0; t < 4; ++t) {
            v2f b = *(const v2f*)(Brow[t] + k + 2 * h);
            acc[t] = wmma4(a, b, acc[t]);
        }
    }

    #pragma unroll
    for (int t = 0; t < 4; ++t) {
        const int col = c0 + t * 16 + r16;
        const float bv = bias[col];
        if (TOUT) {
            // C layout rows m0+8h .. m0+8h+7 are contiguous along v
            v4f lo, hi;
            #pragma unroll
            for (int v = 0; v < 4; ++v) { lo[v] = acc[t][v] + bv; hi[v] = acc[t][v + 4] + bv; }
            float* p = C + (size_t)col * Nrows + m0 + 8 * h;
            *(v4f*)(p)     = lo;
            *(v4f*)(p + 4) = hi;
        } else {
            #pragma unroll
            for (int v = 0; v < 8; ++v) {
                const int row = m0 + v + 8 * h;
                C[(size_t)row * M + col] = acc[t][v] + bv;
            }
        }
    }
}

// ---------------------------------------------------------------------------
// Fused: scores = mask(relu(f @ g^T)); softmax(scores); out = softmax @ xW
// One workgroup per 16-row tile; flash-style sweep over 64-column blocks.
// Phase 1: 8 waves split K=1024 (128 each), partial S tiles via WMMA.
// Phase 2: LDS reduction + relu + adjacency mask.
// Phase 3: online softmax bookkeeping (wave 0, one lane per row).
// Phase 4: each wave rescales its private 16x128 output slice and does
//          P(16x64) @ xW(64x128) with WMMA; B fragments come from the
//          transposed xWT so they are contiguous-along-K b64 loads.
// ---------------------------------------------------------------------------
__global__ __launch_bounds__(256)
void attn_kernel(const float* __restrict__ f, const float* __restrict__ g,
                 const float* __restrict__ xWT /*[O][N]*/, const int* __restrict__ adj,
                 float* __restrict__ out) {
    __shared__ __align__(32) float sp[8][4][32][8];   // per-wave partial S
    __shared__ __align__(16) float Srow[16][68];      // masked scores (padded)
    __shared__ __align__(16) float Prow[16][68];      // exp probabilities
    __shared__ float scaleS[16];
    __shared__ float mst[16];
    __shared__ float lst[16];

    const int tid  = threadIdx.x;
    const int wave = tid >> 5;
    const int lane = tid & 31;
    const int h    = lane >> 4;
    const int r16  = lane & 15;
    const int m0   = blockIdx.x * 16;   // row tile base
    const int kw   = wave * 128;        // this wave's K slice (phase 1)
    const int c0   = wave * 128;        // this wave's O slice (phase 4)

    if (tid < 16) { mst[tid] = -INFINITY; lst[tid] = 0.0f; }

    v8f accPV[8];
    #pragma unroll
    for (int t = 0; t < 8; ++t) accPV[t] = zero8();

    __syncthreads();

    for (int j0 = 0; j0 < Nn; j0 += 64) {
        // ---- Phase 1: partial S = f_tile @ g_block^T over K slice [kw,kw+128)
        v8f accS[4];
        #pragma unroll
        for (int jt = 0; jt < 4; ++jt) accS[jt] = zero8();

        const float* frow = f + (size_t)(m0 + r16) * Dd + kw;
        #pragma unroll 4
        for (int kk = 0; kk < 128; kk += 4) {
            v2f a = *(const v2f*)(frow + kk + 2 * h);
            #pragma unroll
            for (int jt = 0; jt < 4; ++jt) {
                const float* gp = g + (size_t)(j0 + jt * 16 + r16) * Dd + kw + kk + 2 * h;
                v2f b = *(const v2f*)gp;
                accS[jt] = wmma4(a, b, accS[jt]);
            }
        }
        #pragma unroll
        for (int jt = 0; jt < 4; ++jt)
            *(v8f*)&sp[wave][jt][lane][0] = accS[jt];
        __syncthreads();

        // ---- Phase 2: reduce 8 partials, relu + adjacency mask -> Srow
        for (int idx = tid; idx < 1024; idx += 256) {
            const int jt = idx >> 8;
            const int r  = (idx >> 4) & 15;
            const int c  = idx & 15;
            const int pl = c + ((r >= 8) ? 16 : 0);
            const int pv = r & 7;
            float s = 0.0f;
            #pragma unroll
            for (int w = 0; w < 8; ++w) s += sp[w][jt][pl][pv];
            s = fmaxf(s, 0.0f);                                  // relu
            const int av = adj[(size_t)(m0 + r) * Nn + j0 + jt * 16 + c];
            Srow[r][jt * 16 + c] = (av > 0) ? s : -9.0e15f;      // mask
        }
        __syncthreads();

        // ---- Phase 3: online softmax update (wave 0, one lane per row)
        if (wave == 0 && lane < 16) {
            const int r = lane;
            const float mold = mst[r];
            float mnew = mold;
            for (int j = 0; j < 64; ++j) mnew = fmaxf(mnew, Srow[r][j]);
            const float sc = __expf(mold - mnew);   // 0 on first block (mold=-inf)
            float sum = 0.0f;
            for (int j = 0; j < 64; ++j) {
                const float p = __expf(Srow[r][j] - mnew);
                Prow[r][j] = p;
                sum += p;
            }
            lst[r] = lst[r] * sc + sum;
            mst[r] = mnew;
            scaleS[r] = sc;
        }
        __syncthreads();

        // ---- Phase 4: rescale private accumulators, accumulate P @ xW slice
        float sc_r[8];
        #pragma unroll
        for (int v = 0; v < 8; ++v) sc_r[v] = scaleS[v + 8 * h];
        #pragma unroll
        for (int t = 0; t < 8; ++t)
            #pragma unroll
            for (int v = 0; v < 8; ++v) accPV[t][v] *= sc_r[v];

        for (int kk = 0; kk < 64; kk += 4) {
            v2f a = *(v2f*)&Prow[r16][kk + 2 * h];   // ds_load_b64
            #pragma unroll
            for (int t = 0; t < 8; ++t) {
                // B[kb][n] = xW[j0+kb][c0+n] = xWT[c0+n][j0+kb] -> b64 along K
                const float* xp = xWT + (size_t)(c0 + t * 16 + r16) * Nn + j0 + kk + 2 * h;
                v2f b = *(const v2f*)xp;
                accPV[t] = wmma4(a, b, accPV[t]);
            }
        }
        // No barrier needed: next phase-1 writes sp (not read until after the
        // phase-1 barrier), and Prow/scaleS are rewritten only after the next
        // phase-2 barrier, which all waves must reach first.
    }

    __syncthreads();
    // ---- Epilogue: normalize by row sums and store
    float li[8];
    #pragma unroll
    for (int v = 0; v < 8; ++v) li[v] = 1.0f / lst[v + 8 * h];
    #pragma unroll
    for (int t = 0; t < 8; ++t) {
        const int col = c0 + t * 16 + r16;
        #pragma unroll
        for (int v = 0; v < 8; ++v) {
            const int row = m0 + v + 8 * h;
            out[(size_t)row * Oo + col] = accPV[t][v] * li[v];
        }
    }
}

extern "C" void kernel_launch(void* const* d_in, const int* in_sizes, int n_in,
                              void* d_out, int out_size, void* d_ws, size_t ws_size,
                              hipStream_t stream) {
    const float* x   = (const float*)d_in[0];
    const int*   adj = (const int*)  d_in[1];
    const float* Wf  = (const float*)d_in[2];
    const float* bf  = (const float*)d_in[3];
    const float* Wg  = (const float*)d_in[4];
    const float* bg  = (const float*)d_in[5];
    const float* W   = (const float*)d_in[6];
    const float* bW  = (const float*)d_in[7];
    float* out = (float*)d_out;

    float* ws    = (float*)d_ws;
    float* fbuf  = ws;                             // f   [N,D]  32 MB
    float* gbuf  = ws + (size_t)Nn * Dd;           // g   [N,D]  32 MB
    float* xWT   = ws + 2 * (size_t)Nn * Dd;       // xW^T [O,N] 32 MB
    float* wt    = ws + 3 * (size_t)Nn * Dd;       // transposed weight scratch, 4 MB

    dim3 block(256);
    dim3 tgrid(Dd / 64, Dd / 64);
    dim3 ggrid(Nn / 128, Dd / 64);

    // f = x @ Wf + bf        (NT GEMM via pre-transposed weights)
    transpose_kernel<<<tgrid, block, 0, stream>>>(Wf, wt, Dd, Dd);
    gemm_nt_bias_kernel<false><<<ggrid, block, 0, stream>>>(x, wt, bf, fbuf, Nn, Dd, Dd);
    // g = x @ Wg + bg
    transpose_kernel<<<tgrid, block, 0, stream>>>(Wg, wt, Dd, Dd);
    gemm_nt_bias_kernel<false><<<ggrid, block, 0, stream>>>(x, wt, bg, gbuf, Nn, Dd, Dd);
    // xWT = (x @ W + bW)^T   (transposed output for contiguous-K PV loads)
    transpose_kernel<<<tgrid, block, 0, stream>>>(W, wt, Dd, Oo);
    gemm_nt_bias_kernel<true><<<dim3(Nn / 128, Oo / 64), block, 0, stream>>>(x, wt, bW, xWT, Nn, Dd, Oo);

    // Fused masked-softmax attention
    attn_kernel<<<dim3(Nn / 16), block, 0, stream>>>(fbuf, gbuf, xWT, adj, out);
}